// LatentVariable_15444702396648
// MI455X (gfx1250) — compile-verified
//
#include <hip/hip_runtime.h>
#include <hip/hip_bf16.h>
#include <stdint.h>

// z_b = mu[a_b] + tril(L[a_b]) @ eps_b  for B=4.19M rows, A=64, D=8.
// Pure bandwidth-bound (68 B/row, ~12us floor at 23.3 TB/s). Strategy:
//  - stage the 18KB mu/cov table into LDS with the CDNA5 async-to-LDS engine,
//    transposed to slot-major so compute-phase LDS reads map bank == annotator id
//    (distinct a -> distinct bank, equal a -> same address broadcast: conflict-free;
//    compiler fuses the 256B-stride pairs into ds_load_2addr_stride64_b32)
//  - one row per lane, grid-stride; coalesced b128 eps loads / z stores with
//    non-temporal hints (single-use streams; keep L2 for the table).

#define BLOCK 256
#define NA 64
#define ND 8
#define NSLOT (ND + ND * ND) /* 8 mu slots + 64 L slots = 72 */

typedef __attribute__((address_space(1))) int GI; // global (AS1)
typedef __attribute__((address_space(3))) int LI; // LDS (AS3)
typedef __attribute__((ext_vector_type(4))) float v4f;

__global__ __launch_bounds__(BLOCK) void latent_mvn_kernel(
    const int* __restrict__ ann,
    const float* __restrict__ eps,
    const float* __restrict__ mu,
    const float* __restrict__ cov,
    float* __restrict__ out,
    int nrows)
{
    __shared__ float tab[NSLOT * NA]; // 18432 B: tab[slot*64 + a]

    // ---- Stage table into LDS (transposing scatter) via async-to-LDS engine ----
    for (int e = threadIdx.x; e < NSLOT * NA; e += BLOCK) {
        int a = e & (NA - 1);
        int s = e >> 6;
        const float* src = (s < ND) ? (mu + a * ND + s)
                                    : (cov + a * (ND * ND) + (s - ND));
#if __has_builtin(__builtin_amdgcn_global_load_async_to_lds_b32)
        __builtin_amdgcn_global_load_async_to_lds_b32(
            (GI*)(uintptr_t)src,
            (LI*)(uintptr_t)(uint32_t)(uintptr_t)(&tab[e]),
            0, 0);
#else
        tab[e] = *src;
#endif
    }
#if __has_builtin(__builtin_amdgcn_global_load_async_to_lds_b32)
# if __has_builtin(__builtin_amdgcn_s_wait_asynccnt)
    __builtin_amdgcn_s_wait_asynccnt(0);
# else
    asm volatile("s_wait_asynccnt 0" ::: "memory");
# endif
#endif
    __syncthreads();

    // ---- Streaming compute: one row per lane, non-temporal streams ----
    const v4f* __restrict__ eps4 = (const v4f*)eps;
    v4f* __restrict__ out4 = (v4f*)out;

    int tid = blockIdx.x * BLOCK + threadIdx.x;
    int gstride = gridDim.x * BLOCK;
    for (int b = tid; b < nrows; b += gstride) {
        int a = __builtin_nontemporal_load(&ann[b]);
        v4f e0 = __builtin_nontemporal_load(&eps4[2 * b]);
        v4f e1 = __builtin_nontemporal_load(&eps4[2 * b + 1]);
        float e[8] = {e0.x, e0.y, e0.z, e0.w, e1.x, e1.y, e1.z, e1.w};

        float z[8];
#pragma unroll
        for (int j = 0; j < ND; ++j) {
            float acc = tab[j * NA + a]; // mu[a][j], bank == a
#pragma unroll
            for (int k = 0; k <= j; ++k) // lower triangle only (tril)
                acc = __builtin_fmaf(tab[(ND + j * ND + k) * NA + a], e[k], acc);
            z[j] = acc;
        }
        v4f r0 = {z[0], z[1], z[2], z[3]};
        v4f r1 = {z[4], z[5], z[6], z[7]};
        __builtin_nontemporal_store(r0, &out4[2 * b]);
        __builtin_nontemporal_store(r1, &out4[2 * b + 1]);
    }
}

extern "C" void kernel_launch(void* const* d_in, const int* in_sizes, int n_in,
                              void* d_out, int out_size, void* d_ws, size_t ws_size,
                              hipStream_t stream) {
    const int*   ann = (const int*)d_in[0];
    const float* eps = (const float*)d_in[1];
    const float* mu  = (const float*)d_in[2];
    const float* cov = (const float*)d_in[3];
    float* out = (float*)d_out;

    int nrows = in_sizes[0];
    int blocks = (nrows + BLOCK - 1) / BLOCK;
    if (blocks > 2048) blocks = 2048;
    if (blocks < 1) blocks = 1;

    latent_mvn_kernel<<<blocks, BLOCK, 0, stream>>>(ann, eps, mu, cov, out, nrows);
}